// BasicDeconvolutionBlock4d_4063039062842
// MI455X (gfx1250) — compile-verified
//
#include <hip/hip_runtime.h>

typedef float v2f __attribute__((ext_vector_type(2)));
typedef float v8f __attribute__((ext_vector_type(8)));

#define INC   64
#define OUTC  96
#define NTILE 6          // OUTC / 16
#define NOFF  26
#define EPS   1e-5f

// LDS weight layout: pair-interleaved, row = K-pair (k/2), col = output channel.
// Wlds[k2 * WSTRIDE + c] = { W[2*k2][c], W[2*k2+1][c] }  (float2)
// WSTRIDE = 112 so lanes 0-15 (k2 = k/2) and lanes 16-31 (k2 = k/2+1) hit
// complementary 32-bank halves on b64 loads (2*112 mod 64 == 32): no conflicts.
#define WSTRIDE 112
#define WLDS_N  ((INC / 2) * WSTRIDE)     // 3584 float2 = 28 KB

// ---------------------------------------------------------------------------
// WMMA inner loop: accumulate a 16(M) x 96(N) f32 tile, K = 64, via
// V_WMMA_F32_16X16X4_F32 (exact f32 math, matches reference numerics).
// A (16x4 f32): lanes 0-15 hold K={k,k+1}, lanes 16-31 hold K={k+2,k+3}, M=lane&15.
// B (4x16 f32): one ds_load_b64 per lane from the pair-interleaved LDS tile.
// ---------------------------------------------------------------------------
__device__ __forceinline__ void wmma_16x96_k64(const float* __restrict__ frow,
                                               bool rowOK,
                                               const v2f* __restrict__ Wlds,
                                               int half, int l15,
                                               v8f acc[NTILE]) {
#pragma unroll
  for (int k = 0; k < INC; k += 4) {
    // per-lane pair of consecutive K values (8B aligned: k even, row 256B aligned)
    v2f a = *reinterpret_cast<const v2f*>(frow + k + half * 2);
    if (!rowOK) { a.x = 0.0f; a.y = 0.0f; }
    const int k2 = (k >> 1) + half;                  // K-pair row for this lane half
#pragma unroll
    for (int t = 0; t < NTILE; ++t) {
      const v2f b = Wlds[k2 * WSTRIDE + t * 16 + l15];   // single ds_load_b64
      acc[t] = __builtin_amdgcn_wmma_f32_16x16x4_f32(
          /*neg_a=*/false, a, /*neg_b=*/false, b,
          /*c_mod=*/(short)0, acc[t], /*reuse_a=*/false, /*reuse_b=*/false);
    }
  }
}

// Cooperative stage of one 64x96 weight tile (row-major in global) into the
// pair-interleaved, padded LDS layout. 3072 elements / 256 threads.
__device__ __forceinline__ void stage_weights(const float* __restrict__ W,
                                              v2f* __restrict__ Wlds) {
  for (int i = threadIdx.x; i < (INC / 2) * OUTC; i += 256) {
    const int k2 = i / OUTC;
    const int c  = i - k2 * OUTC;
    v2f w;
    w.x = W[(2 * k2 + 0) * OUTC + c];
    w.y = W[(2 * k2 + 1) * OUTC + c];
    Wlds[k2 * WSTRIDE + c] = w;
  }
}

// ---------------------------------------------------------------------------
// Kernel 1: out[n, :] = feats[n, :] @ W_center   (initializes all of d_out)
// ---------------------------------------------------------------------------
__global__ void __launch_bounds__(256)
center_gemm_kernel(const float* __restrict__ feats, const float* __restrict__ Wc,
                   float* __restrict__ out, int N) {
  __shared__ v2f Wlds[WLDS_N];                       // 28 KB
  stage_weights(Wc, Wlds);
  __syncthreads();

  const int lane = threadIdx.x & 31;
  const int tile = blockIdx.x * 8 + (threadIdx.x >> 5);
  const int row0 = tile * 16;
  if (row0 >= N) return;                             // wave-uniform exit

  const int half = lane >> 4, l15 = lane & 15;
  const int arow  = row0 + l15;
  const bool rowOK = (arow < N);
  const float* frow = feats + (size_t)(rowOK ? arow : (N - 1)) * INC;

  v8f acc[NTILE] = {};
  wmma_16x96_k64(frow, rowOK, Wlds, half, l15, acc);

  // C layout: VGPR r -> M = r (lanes 0-15) / M = 8+r (lanes 16-31)
  const int mbase = half * 8;
#pragma unroll
  for (int t = 0; t < NTILE; ++t) {
#pragma unroll
    for (int r = 0; r < 8; ++r) {
      const int orow = row0 + mbase + r;
      if (orow < N) out[(size_t)orow * OUTC + t * 16 + l15] = acc[t][r];
    }
  }
}

// ---------------------------------------------------------------------------
// Kernel 2: per offset k: gather feats[in_idx], GEMM vs W_off[k],
//           scatter-add into out via out_idx (sink row N dropped).
// grid = (26, ceil(PT/8)), block = 256: all 8 waves share one staged weight tile.
// ---------------------------------------------------------------------------
__global__ void __launch_bounds__(256)
offset_gemm_scatter_kernel(const float* __restrict__ feats,
                           const float* __restrict__ Woff,
                           const int* __restrict__ in_idx,
                           const int* __restrict__ out_idx,
                           float* __restrict__ out, int N, int P) {
  __shared__ v2f Wlds[WLDS_N];                       // 28 KB
  const int koff = blockIdx.x;                       // 0..25
  stage_weights(Woff + (size_t)koff * INC * OUTC, Wlds);
  __syncthreads();

  const int lane  = threadIdx.x & 31;
  const int ptile = blockIdx.y * 8 + (threadIdx.x >> 5);
  const int p0    = ptile * 16;
  if (p0 >= P) return;                               // wave-uniform exit

  const int half = lane >> 4, l15 = lane & 15;
  const int* inK  = in_idx  + (size_t)koff * P;
  const int* outK = out_idx + (size_t)koff * P;

  const int p  = p0 + l15;
  const int pc = (p < P) ? p : (P - 1);              // clamp tail loads (harmless)
  const int grow = inK[pc];                          // gather row (always >= 0)
  const float* frow = feats + (size_t)grow * INC;

  v8f acc[NTILE] = {};
  wmma_16x96_k64(frow, true, Wlds, half, l15, acc);

  const int mbase = half * 8;
  int orow[8];
#pragma unroll
  for (int r = 0; r < 8; ++r) {
    const int pp = p0 + mbase + r;
    orow[r] = (pp < P) ? outK[pp] : N;               // N == sink -> dropped below
  }
#pragma unroll
  for (int t = 0; t < NTILE; ++t) {
#pragma unroll
    for (int r = 0; r < 8; ++r) {
      if (orow[r] < N)
        atomicAdd(&out[(size_t)orow[r] * OUTC + t * 16 + l15], acc[t][r]);
    }
  }
}

// ---------------------------------------------------------------------------
// BatchNorm reduction: ws[0..95] = sum, ws[96..191] = sum of squares
// ---------------------------------------------------------------------------
__global__ void zero_ws_kernel(float* __restrict__ ws) {
  const int i = threadIdx.x;
  if (i < 2 * OUTC) ws[i] = 0.0f;
}

__global__ void __launch_bounds__(384)
bn_reduce_kernel(const float* __restrict__ out, float* __restrict__ ws, int N) {
  const int c  = threadIdx.x;                        // 0..95 -> coalesced per row
  const int ty = threadIdx.y;                        // 0..3
  float s = 0.0f, s2 = 0.0f;
  for (int r = blockIdx.x * 4 + ty; r < N; r += gridDim.x * 4) {
    const float v = out[(size_t)r * OUTC + c];
    s += v;
    s2 += v * v;
  }
  atomicAdd(&ws[c], s);
  atomicAdd(&ws[OUTC + c], s2);
}

// ---------------------------------------------------------------------------
// BN apply + ReLU (in place)
// ---------------------------------------------------------------------------
__global__ void __launch_bounds__(256)
bn_apply_kernel(float* __restrict__ out, const float* __restrict__ ws,
                const float* __restrict__ gamma, const float* __restrict__ beta,
                int N) {
  const size_t e = (size_t)blockIdx.x * 256 + threadIdx.x;
  const size_t total = (size_t)N * OUTC;
  if (e >= total) return;
  const int c = (int)(e % OUTC);
  const float invN = 1.0f / (float)N;
  const float mean = ws[c] * invN;
  const float var  = ws[OUTC + c] * invN - mean * mean;
  const float inv  = rsqrtf(var + EPS);
  const float y = (out[e] - mean) * inv * gamma[c] + beta[c];
  out[e] = fmaxf(y, 0.0f);
}

// ---------------------------------------------------------------------------
extern "C" void kernel_launch(void* const* d_in, const int* in_sizes, int n_in,
                              void* d_out, int out_size, void* d_ws, size_t ws_size,
                              hipStream_t stream) {
  const float* feats = (const float*)d_in[0];
  const float* Wc    = (const float*)d_in[1];
  const float* Woff  = (const float*)d_in[2];
  const float* gamma = (const float*)d_in[3];
  const float* beta  = (const float*)d_in[4];
  const int*   inI   = (const int*)d_in[5];
  const int*   outI  = (const int*)d_in[6];
  float* out = (float*)d_out;
  float* ws  = (float*)d_ws;

  const int N = in_sizes[0] / INC;
  const int P = in_sizes[5] / NOFF;

  // 1) center GEMM (fully initializes d_out)
  const int RT = (N + 15) / 16;
  center_gemm_kernel<<<(RT + 7) / 8, 256, 0, stream>>>(feats, Wc, out, N);

  // 2) 26 offset GEMMs + scatter-add
  const int PT = (P + 15) / 16;
  dim3 g2(NOFF, (PT + 7) / 8);
  offset_gemm_scatter_kernel<<<g2, 256, 0, stream>>>(feats, Woff, inI, outI, out, N, P);

  // 3) BN statistics
  zero_ws_kernel<<<1, 256, 0, stream>>>(ws);
  dim3 b3(OUTC, 4);
  bn_reduce_kernel<<<768, b3, 0, stream>>>(out, ws, N);

  // 4) normalize + ReLU
  const size_t total = (size_t)N * OUTC;
  bn_apply_kernel<<<(unsigned)((total + 255) / 256), 256, 0, stream>>>(out, ws, gamma, beta, N);
}